// CLIPTrainLoss_59974923321751
// MI455X (gfx1250) — compile-verified
//
#include <hip/hip_runtime.h>
#include <math.h>

typedef float v2f __attribute__((ext_vector_type(2)));
typedef float v8f __attribute__((ext_vector_type(8)));

#define BATCH   32
#define MROWS   120
#define FRAMES  64
#define DDIM    512
#define MPAD    128

__device__ __forceinline__ float blockReduceSum(float v, float* red, int tid, int nthreads) {
    red[tid] = v;
    __syncthreads();
    for (int off = nthreads >> 1; off > 0; off >>= 1) {
        if (tid < off) red[tid] += red[tid + off];
        __syncthreads();
    }
    return red[0];
}

// ---------------------------------------------------------------------------
// Kernel 1: skeleton mean over 64 frames + L2 normalize.
// One block per (b, n) row; 256 threads, each owns 2 of the 512 dims.
// Streams 128 KB per block, fully coalesced float2 -> global_load_b64.
// This kernel is the HBM-bandwidth floor (~503 MB @ 23.3 TB/s ~ 22 us).
// ---------------------------------------------------------------------------
__global__ void skel_mean_norm_kernel(const float* __restrict__ skel,
                                      float* __restrict__ skel_f) {
    const int row = blockIdx.x;          // b*120 + n
    const int tid = threadIdx.x;         // 0..255
    const float* __restrict__ base = skel + (size_t)row * FRAMES * DDIM + tid * 2;

    float s0 = 0.0f, s1 = 0.0f;
    #pragma unroll 8
    for (int f = 0; f < FRAMES; ++f) {
        const float2 v = *(const float2*)(base + (size_t)f * DDIM);
        s0 += v.x;
        s1 += v.y;
    }
    s0 *= (1.0f / FRAMES);
    s1 *= (1.0f / FRAMES);

    __shared__ float red[256];
    const float tot = blockReduceSum(s0 * s0 + s1 * s1, red, tid, 256);
    const float rn = 1.0f / sqrtf(tot);

    float2 o;
    o.x = s0 * rn;
    o.y = s1 * rn;
    *(float2*)(skel_f + (size_t)row * DDIM + tid * 2) = o;
}

// ---------------------------------------------------------------------------
// Kernel 2: text normalize, with LOGIT_SCALE folded in.
// ---------------------------------------------------------------------------
__global__ void text_norm_kernel(const float* __restrict__ text,
                                 float* __restrict__ text_f) {
    const int row = blockIdx.x;          // b*120 + m
    const int tid = threadIdx.x;         // 0..255
    const float2 v = *(const float2*)(text + (size_t)row * DDIM + tid * 2);

    __shared__ float red[256];
    const float tot = blockReduceSum(v.x * v.x + v.y * v.y, red, tid, 256);
    const float s = 14.285714285714286f / sqrtf(tot);   // LOGIT_SCALE / ||t||

    float2 o;
    o.x = v.x * s;
    o.y = v.y * s;
    *(float2*)(text_f + (size_t)row * DDIM + tid * 2) = o;
}

// ---------------------------------------------------------------------------
// Kernel 3: batched GEMM via V_WMMA_F32_16X16X4_F32 (full f32 precision).
// Grid: 32 batches x 8 m-tiles = 256 blocks; 8 waves/block, one 16x16
// n-tile per wave. K-loop over 512 in steps of 8 with two accumulators
// to break the WMMA RAW chain. Operands live in L2 (2 x 7.9 MB).
//
// A-matrix 16x4 f32 layout: lane L -> M = L%16, VGPR j -> K = j + 2*(L/16).
// B-matrix 4x16 mirrors it (B[k][n] = skel_f[n][k]), so both operands are
// the same float2 gather pattern from the row-major normalized embeddings.
// Tile rows beyond 120 are clamped to row 0: the garbage results land only
// in the padded region of the 128x128 logits buffer, which kernel 4 never
// reads, so EXEC stays all-ones (WMMA requirement) with zero masking cost.
// ---------------------------------------------------------------------------
__global__ void wmma_logits_kernel(const float* __restrict__ text_f,
                                   const float* __restrict__ skel_f,
                                   float* __restrict__ logits) {
    const int b    = blockIdx.x >> 3;
    const int mt   = blockIdx.x & 7;
    const int wave = threadIdx.x >> 5;      // n-tile index 0..7
    const int lane = threadIdx.x & 31;
    const int r16  = lane & 15;
    const int kq   = (lane >> 4) << 1;      // 0 for lanes 0-15, 2 for lanes 16-31

    int m = mt * 16 + r16;
    int n = wave * 16 + r16;
    if (m >= MROWS) m = 0;                  // clamp; padded output ignored later
    if (n >= MROWS) n = 0;

    const float* __restrict__ Ap = text_f + ((size_t)b * MROWS + m) * DDIM + kq;
    const float* __restrict__ Bp = skel_f + ((size_t)b * MROWS + n) * DDIM + kq;

    v8f acc0 = {};
    v8f acc1 = {};
    for (int k = 0; k < DDIM; k += 8) {
        const float2 a0 = *(const float2*)(Ap + k);
        const float2 b0 = *(const float2*)(Bp + k);
        const float2 a1 = *(const float2*)(Ap + k + 4);
        const float2 b1 = *(const float2*)(Bp + k + 4);
        v2f av0; av0.x = a0.x; av0.y = a0.y;
        v2f bv0; bv0.x = b0.x; bv0.y = b0.y;
        v2f av1; av1.x = a1.x; av1.y = a1.y;
        v2f bv1; bv1.x = b1.x; bv1.y = b1.y;
        acc0 = __builtin_amdgcn_wmma_f32_16x16x4_f32(
            false, av0, false, bv0, (short)0, acc0, false, false);
        acc1 = __builtin_amdgcn_wmma_f32_16x16x4_f32(
            false, av1, false, bv1, (short)0, acc1, false, false);
    }

    // D layout: VGPR v -> row (v + 8*(lane/16)), col = lane%16 within the tile.
    float* __restrict__ out =
        logits + (size_t)b * MPAD * MPAD + (size_t)(mt * 16) * MPAD + wave * 16;
    const int rowoff = (lane >> 4) * 8;
    const int col = lane & 15;
    #pragma unroll
    for (int v = 0; v < 8; ++v) {
        out[(size_t)(v + rowoff) * MPAD + col] = acc0[v] + acc1[v];
    }
}

// ---------------------------------------------------------------------------
// Kernel 4: per-batch sum over rows of (diag - logsumexp(row)).
// One block per batch, thread i owns row i (reads only n < 120 — padded
// columns/rows of the logits buffer are never touched). Fixed-order LDS
// reduction keeps the result deterministic.
// ---------------------------------------------------------------------------
__global__ void lse_diag_kernel(const float* __restrict__ logits,
                                float* __restrict__ partials) {
    const int b = blockIdx.x;
    const int tid = threadIdx.x;            // 0..127
    float val = 0.0f;
    if (tid < MROWS) {
        const float* __restrict__ row =
            logits + (size_t)b * MPAD * MPAD + (size_t)tid * MPAD;
        float mx = -INFINITY;
        for (int n = 0; n < MROWS; ++n) mx = fmaxf(mx, row[n]);
        float s = 0.0f;
        for (int n = 0; n < MROWS; ++n) s += expf(row[n] - mx);
        const float lse = mx + logf(s);
        val = row[tid] - lse;               // logp[b, tid, tid]
    }
    __shared__ float red[128];
    const float tot = blockReduceSum(val, red, tid, 128);
    if (tid == 0) partials[b] = tot;
}

__global__ void finalize_kernel(const float* __restrict__ partials,
                                float* __restrict__ out) {
    if (threadIdx.x == 0 && blockIdx.x == 0) {
        float s = 0.0f;
        for (int i = 0; i < BATCH; ++i) s += partials[i];
        out[0] = -s / (float)(BATCH * MROWS);
    }
}

// ---------------------------------------------------------------------------
// Workspace layout (floats):
//   text_f   : [32*120*512]            = 1,966,080
//   skel_f   : [32*120*512]            = 1,966,080
//   logits   : [32*128*128] (padded)   =   524,288
//   partials : [32]
// Total ~ 17.8 MB.
// ---------------------------------------------------------------------------
extern "C" void kernel_launch(void* const* d_in, const int* in_sizes, int n_in,
                              void* d_out, int out_size, void* d_ws, size_t ws_size,
                              hipStream_t stream) {
    const float* skel = (const float*)d_in[0];   // (32,120,64,512) f32
    const float* text = (const float*)d_in[1];   // (32,120,512)    f32

    float* ws       = (float*)d_ws;
    float* text_f   = ws;
    float* skel_f   = ws + (size_t)BATCH * MROWS * DDIM;
    float* logits   = skel_f + (size_t)BATCH * MROWS * DDIM;
    float* partials = logits + (size_t)BATCH * MPAD * MPAD;

    skel_mean_norm_kernel<<<BATCH * MROWS, 256, 0, stream>>>(skel, skel_f);
    text_norm_kernel<<<BATCH * MROWS, 256, 0, stream>>>(text, text_f);
    wmma_logits_kernel<<<BATCH * 8, 256, 0, stream>>>(text_f, skel_f, logits);
    lse_diag_kernel<<<BATCH, 128, 0, stream>>>(logits, partials);
    finalize_kernel<<<1, 32, 0, stream>>>(partials, (float*)d_out);
}